// QuantumFilter_65481071407377
// MI455X (gfx1250) — compile-verified
//
#include <hip/hip_runtime.h>

// CDNA5 / gfx1250. Output = 205.5 MB of the repeating 16-byte pattern
// {v0,v1,v2,v3}; pure store-bandwidth, floor ~8.8us @ 23.3 TB/s.
// Input x (65536x784) is mathematically ignored -> never loaded.
//
// Two kernels:
//  1) 1-wave setup: computes vec[4] = {cos(ry0)*cos(sum rl), cos(ry1..3)}
//     into d_ws (keeps the big ocml-cos code out of the streaming kernel).
//  2) streaming kernel: uniform s_load of vec, one V_WMMA_F32_16X16X4_F32
//     (A[m][0]=vec[m%4], B[0][n]=1 => D[m][n]=vec[m%4]) materializes
//     d[0..3]=d[4..7]={v0,v1,v2,v3} per lane, then 8 unguarded
//     non-temporal global_store_b128 per thread (512B contiguous per wave
//     per store).

typedef __attribute__((ext_vector_type(2))) float v2f;
typedef __attribute__((ext_vector_type(4))) float f32x4;
typedef __attribute__((ext_vector_type(8))) float v8f;

#define THREADS 256
#define F4_PER_THREAD 8
#define F4_PER_BLOCK (THREADS * F4_PER_THREAD)  // 2048

__global__ __launch_bounds__(32) void qf_setup_kernel(
    const float* __restrict__ ry, const float* __restrict__ rl,
    float* __restrict__ vec) {
  if (threadIdx.x == 0) {
    float s = rl[0] + rl[1] + rl[2] + rl[3] + rl[4] + rl[5] + rl[6] + rl[7];
    vec[0] = cosf(ry[0]) * cosf(s);
    vec[1] = cosf(ry[1]);
    vec[2] = cosf(ry[2]);
    vec[3] = cosf(ry[3]);
  }
}

__global__ __launch_bounds__(THREADS) void qf_store_kernel(
    const float* __restrict__ vec, f32x4* __restrict__ out4, int n4) {
  // Uniform address -> scalar load (s_load_b128) of the 4-float pattern.
  float v0 = vec[0], v1 = vec[1], v2 = vec[2], v3 = vec[3];

  // WMMA operands per the 16x4 (A) / 4x16 (B) f32 VGPR layouts:
  // lanes 0-15 hold K=0 in VGPR0; only K=0 is nonzero.
  int lane = threadIdx.x & 31;  // wave32
  int m4 = lane & 3;
  float vm = (m4 == 0) ? v0 : (m4 == 1) ? v1 : (m4 == 2) ? v2 : v3;

  v2f a, b;
  a.x = (lane < 16) ? vm : 0.0f;
  a.y = 0.0f;
  b.x = (lane < 16) ? 1.0f : 0.0f;
  b.y = 0.0f;
  v8f c = {};
  // D[m][n] = A[m][0]*B[0][n] = vec[m%4]; EXEC is all-1s here (no prior
  // divergence). 8 args: (neg_a, A, neg_b, B, c_mod, C, reuse_a, reuse_b)
  v8f d = __builtin_amdgcn_wmma_f32_16x16x4_f32(
      false, a, false, b, (short)0, c, false, false);

  // C/D layout: lane L VGPR r holds M = r + 8*(L/16); M%4 == r%4, so every
  // lane gets d[0..3] = d[4..7] = {v0,v1,v2,v3} -- the 16B store pattern.
  f32x4 val0 = __builtin_shufflevector(d, d, 0, 1, 2, 3);
  f32x4 val1 = __builtin_shufflevector(d, d, 4, 5, 6, 7);

  int base = blockIdx.x * F4_PER_BLOCK + threadIdx.x;
  if (base + (F4_PER_THREAD - 1) * THREADS < n4) {
    // Fast path (every block when n4 % 2048 == 0): 8 straight-line
    // non-temporal b128 stores, no EXEC juggling, clause-friendly.
#pragma unroll
    for (int j = 0; j < F4_PER_THREAD; j += 2) {
      __builtin_nontemporal_store(val0, &out4[base + j * THREADS]);
      __builtin_nontemporal_store(val1, &out4[base + (j + 1) * THREADS]);
    }
  } else {
#pragma unroll
    for (int j = 0; j < F4_PER_THREAD; j += 2) {
      int i0 = base + j * THREADS;
      int i1 = base + (j + 1) * THREADS;
      if (i0 < n4) out4[i0] = val0;
      if (i1 < n4) out4[i1] = val1;
    }
  }
}

extern "C" void kernel_launch(void* const* d_in, const int* in_sizes, int n_in,
                              void* d_out, int out_size, void* d_ws, size_t ws_size,
                              hipStream_t stream) {
  // setup_inputs order: x (unused), ry[4], rl[8]
  const float* ry = (const float*)d_in[1];
  const float* rl = (const float*)d_in[2];
  float* vec = (float*)d_ws;           // 16 bytes of scratch
  f32x4* out4 = (f32x4*)d_out;

  int n4 = out_size / 4;                                // 12,845,056 float4s
  int blocks = (n4 + F4_PER_BLOCK - 1) / F4_PER_BLOCK;  // 6272 exactly

  qf_setup_kernel<<<1, 32, 0, stream>>>(ry, rl, vec);
  qf_store_kernel<<<blocks, THREADS, 0, stream>>>(vec, out4, n4);
}